// SiameseOutputLayer_41644002902557
// MI455X (gfx1250) — compile-verified
//
#include <hip/hip_runtime.h>

// out[n,m] = exp(-||x_n - y_m||^2 / 2) = exp( x_n . y_m - 0.5*(||x_n||^2 + ||y_m||^2) )
// N = M = 4096, D = 32, fp32 in / fp32 out.
//
// MI455X roofline: output = 64 MB write-once @ 23.3 TB/s => ~2.9 us floor; the
// GEMM core is only ~1 GFLOP, so the kernel is store-bound. Design: single pass,
// f32 WMMA (16x16x4, 8 chained for K=32) for the cross term, norms fused into
// the fragment loads, fast v_exp_f32 epilogue, non-temporal stores.
//
// Round-2 changes vs round 1 (from asm inspection):
//  - Hoisted the 8 ds_bpermute row-norm broadcasts (loop-invariant) out of the
//    16-iteration tile loop, pre-scaled as bias[v] = -0.5*na[m].
//  - #pragma unroll 2 on the tile loop so the scheduler can pipeline B-tile
//    loads of one iteration against WMMA + epilogue of the other.

typedef float v2f __attribute__((ext_vector_type(2)));
typedef float v8f __attribute__((ext_vector_type(8)));

#define TILE 16
#define TN_PER_WAVE 16
#define WAVES_PER_BLOCK 8

__global__ __launch_bounds__(256) void rbf_wmma_f32_kernel(
    const float* __restrict__ X,   // [N, 32] row-major
    const float* __restrict__ Y,   // [M, 32] row-major
    float* __restrict__ out,       // [N, M]
    int N, int M)
{
    const int lane      = threadIdx.x & 31;
    const int waveInBlk = threadIdx.x >> 5;
    const int gw        = blockIdx.x * WAVES_PER_BLOCK + waveInBlk;

    const int groups = (M / TILE) / TN_PER_WAVE;     // tn-groups per tm row
    const int tm     = gw / groups;
    const int tnBase = (gw % groups) * TN_PER_WAVE;

    // f32 16x16x4 A-operand layout: lane L holds row M = L%16;
    // VGPR pair covers K = (L/16)*2 + {0,1}; fragments step K by 4.
    const int half = lane >> 4;
    const int r    = lane & 15;

    // ---- A fragments for this wave's 16x32 X tile, row-norm fused ----
    const float* xrow = X + (size_t)(tm * TILE + r) * 32 + half * 2;
    v2f a[8];
    float na = 0.0f;
#pragma unroll
    for (int kk = 0; kk < 8; ++kk) {
        a[kk] = *(const v2f*)(xrow + kk * 4);
        na += a[kk].x * a[kk].x + a[kk].y * a[kk].y;
    }
    na += __shfl_xor(na, 16, 32);   // full ||x_{tm*16+r}||^2 in every lane

    // ---- Hoisted epilogue bias: bias[v] = -0.5 * ||x_{tm*16 + m}||^2 ----
    // C/D layout: lane holds column n = r; VGPR v holds row m = v + 8*half.
    float bias[8];
#pragma unroll
    for (int v = 0; v < 8; ++v)
        bias[v] = -0.5f * __shfl(na, v + half * 8, 32);

    const float* ybase = Y + (size_t)(tnBase * TILE + r) * 32 + half * 2;
    float* obase = out + (size_t)(tm * TILE) * M + (size_t)tnBase * TILE + r;

#pragma unroll 2
    for (int t = 0; t < TN_PER_WAVE; ++t) {
        // ---- B fragments (16x32 Y tile used as B = Y^T), norm fused ----
        const float* yrow = ybase + (size_t)t * TILE * 32;
        v2f b[8];
#pragma unroll
        for (int kk = 0; kk < 8; ++kk)
            b[kk] = *(const v2f*)(yrow + kk * 4);

        float nb = 0.0f;
#pragma unroll
        for (int kk = 0; kk < 8; ++kk)
            nb += b[kk].x * b[kk].x + b[kk].y * b[kk].y;
        nb += __shfl_xor(nb, 16, 32);   // full ||y_{tn*16+r}||^2
        const float nbh = 0.5f * nb;

        // ---- Cross term: 8 chained f32 WMMAs (K = 32) ----
        v8f c = {0.f, 0.f, 0.f, 0.f, 0.f, 0.f, 0.f, 0.f};
#pragma unroll
        for (int kk = 0; kk < 8; ++kk) {
            // 8 args: (neg_a, A, neg_b, B, c_mod, C, reuse_a, reuse_b)
            c = __builtin_amdgcn_wmma_f32_16x16x4_f32(
                    false, a[kk], false, b[kk], (short)0, c, false, false);
        }

        // ---- Epilogue: exp(dot - 0.5*(na+nb)), non-temporal store ----
        float* op = obase + (size_t)t * TILE;
#pragma unroll
        for (int v = 0; v < 8; ++v) {
            const float val = __expf(c[v] + (bias[v] - nbh));
            __builtin_nontemporal_store(val, op + (size_t)(v + half * 8) * M);
        }
    }
}

extern "C" void kernel_launch(void* const* d_in, const int* in_sizes, int n_in,
                              void* d_out, int out_size, void* d_ws, size_t ws_size,
                              hipStream_t stream) {
    const float* i1 = (const float*)d_in[0];   // [1, N, 32]
    const float* i2 = (const float*)d_in[1];   // [1, M, 32]
    float* out = (float*)d_out;                // [1, N, M]

    const int D = 32;
    const int N = in_sizes[0] / D;   // 4096
    const int M = in_sizes[1] / D;   // 4096

    const int tilesM = N / TILE;                      // 256
    const int groups = (M / TILE) / TN_PER_WAVE;      // 16
    const int totalWaves = tilesM * groups;           // 4096
    const int blocks = totalWaves / WAVES_PER_BLOCK;  // 512

    rbf_wmma_f32_kernel<<<blocks, WAVES_PER_BLOCK * 32, 0, stream>>>(i1, i2, out, N, M);
}